// STCeptionLayer_max_17557826306671
// MI455X (gfx1250) — compile-verified
//
#include <hip/hip_runtime.h>
#include <hip/hip_bf16.h>
#include <math.h>

// ---------------------------------------------------------------------------
// STCeption layer for MI455X (gfx1250).
// Roofline: ~1 GB total traffic / 23.3 TB/s => memory bound (~45us); matmul
// FLOPs (~2 GFLOP) are negligible vs matrix-pipe peak, so we keep full fp32
// precision and use V_WMMA_F32_16X16X4_F32 for every channel-mixing GEMM
// (compress / maxproj / einsum / expand). wave32, 8 waves per 256-thr block.
// The shared W tile is staged into LDS with GLOBAL_LOAD_ASYNC_TO_LDS_B32
// (ASYNCcnt path, no VGPR round-trip), synchronized via s_wait_asynccnt.
// ---------------------------------------------------------------------------

typedef __attribute__((ext_vector_type(2))) float v2f;
typedef __attribute__((ext_vector_type(8))) float v8f;

#define EPSF 1e-5f
#define Bv   8
#define Cx   256
#define CHm  64
#define Dv   16
#define Hv   28
#define Wvv  28
#define NPOS (Dv * Hv * Wvv)   // 12544

// ---------------------------------------------------------------------------
// Generic fp32 WMMA GEMM over spatial positions:
//   Y[m, n] = sum_k W[m, k] * X[k, n]   (channel stride == N, NCDHW layout)
// blockIdx.z = batch, blockIdx.y = group; strides passed per call.
// Optional epilogue (expand): Y2 = Y + RES (residual add producing `out`).
// ---------------------------------------------------------------------------
template <int KVAL>
__global__ __launch_bounds__(256) void gemm_wmma_f32(
    const float* __restrict__ X, const float* __restrict__ Wm,
    float* __restrict__ Y, float* __restrict__ Y2, const float* __restrict__ RES,
    int M, int N,
    long xB, long wB, long yB,
    long xG, long wG, long yG)
{
    __shared__ float Wlds[64 * 64];

    const int b = blockIdx.z;
    const int g = blockIdx.y;
    const float* Xp  = X  + (long)b * xB + (long)g * xG;
    const float* Wp  = Wm + (long)b * wB + (long)g * wG;
    float*       Yp  = Y  + (long)b * yB + (long)g * yG;
    float*       Y2p = Y2  ? Y2  + (long)b * yB + (long)g * yG : nullptr;
    const float* Rp  = RES ? RES + (long)b * yB + (long)g * yG : nullptr;

    const int tid = threadIdx.x;

    // Stage the (reused-by-all-waves) W tile into LDS via the CDNA5 async
    // copy path: global memory copied straight to LDS, tracked by ASYNCcnt. The LDS byte
    // address comes from casting the AS(3) pointer; global address in a
    // VGPR pair (GV mode, SADDR=off).
    for (int i = tid; i < M * KVAL; i += 256) {
        const unsigned ldsOff = (unsigned)(uintptr_t)(&Wlds[i]);
        const float* gp = Wp + i;
        asm volatile("global_load_async_to_lds_b32 %0, %1, off"
                     :: "v"(ldsOff), "v"(gp)
                     : "memory");
    }
    asm volatile("s_wait_asynccnt 0" ::: "memory");
    __syncthreads();

    const int lane  = tid & 31;
    const int wave  = tid >> 5;
    const int n0    = (blockIdx.x * 8 + wave) * 16;
    if (n0 >= N) return;                       // wave-uniform; EXEC stays full
    const int nc    = n0 + (lane & 15);
    const int khalf = (lane >> 4) * 2;         // lanes 16-31 carry K+2,K+3

    // Streamed operand: hint next tile into GL2 (global_prefetch_b8).
    __builtin_prefetch(Xp + nc + 128, 0, 1);

    // Load all B-fragments of X once; reuse across the M tiles.
    constexpr int KS = KVAL / 4;
    v2f bfrag[KS];
#pragma unroll
    for (int s = 0; s < KS; ++s) {
        const int k = s * 4 + khalf;
        bfrag[s].x = Xp[(long)k * N + nc];
        bfrag[s].y = Xp[(long)(k + 1) * N + nc];
    }

    for (int m0 = 0; m0 < M; m0 += 16) {
        v8f acc = {};
        const int ma = m0 + (lane & 15);
#pragma unroll
        for (int s = 0; s < KS; ++s) {
            v2f a;
            a.x = Wlds[ma * KVAL + s * 4 + khalf];
            a.y = Wlds[ma * KVAL + s * 4 + khalf + 1];
            acc = __builtin_amdgcn_wmma_f32_16x16x4_f32(
                false, a, false, bfrag[s], (short)0, acc, false, false);
        }
        const int mbase = m0 + ((lane >> 4) * 8);
#pragma unroll
        for (int r = 0; r < 8; ++r) {
            const long idx = (long)(mbase + r) * N + nc;
            const float v = acc[r];
            Yp[idx] = v;
            if (Y2p) Y2p[idx] = v + Rp[idx];
        }
    }
}

// ---------------------------------------------------------------------------
// Naive 3D conv (tower + temporal/spatial branch convs; all tiny FLOPs).
// Channel-window support via inC0/inCT, outC0/outCT for the 21/21/22 splits.
// ---------------------------------------------------------------------------
__global__ void conv3d_naive(
    const float* __restrict__ in, const float* __restrict__ wgt,
    const float* __restrict__ bias, float* __restrict__ out,
    int Cin, int Di, int Hi, int Wi,
    int Cout, int Do, int Ho, int Wo,
    int kd, int kh, int kw, int sd, int sh, int sw,
    int pd, int ph, int pw, int dd, int dh, int dw,
    int inC0, int inCT, int outC0, int outCT)
{
    const long total = (long)Bv * Cout * Do * Ho * Wo;
    for (long i = blockIdx.x * (long)blockDim.x + threadIdx.x; i < total;
         i += (long)gridDim.x * blockDim.x) {
        int w = (int)(i % Wo); long t = i / Wo;
        int h = (int)(t % Ho); t /= Ho;
        int d = (int)(t % Do); t /= Do;
        int o = (int)(t % Cout); int b = (int)(t / Cout);
        float acc = bias ? bias[o] : 0.f;
        for (int ci = 0; ci < Cin; ++ci) {
            const float* ip = in + ((long)b * inCT + inC0 + ci) * (long)Di * Hi * Wi;
            const float* wp = wgt + (((long)o * Cin + ci) * kd) * kh * kw;
            for (int z = 0; z < kd; ++z) {
                const int zi = d * sd - pd + z * dd;
                if (zi < 0 || zi >= Di) continue;
                for (int y = 0; y < kh; ++y) {
                    const int yi = h * sh - ph + y * dh;
                    if (yi < 0 || yi >= Hi) continue;
                    for (int x = 0; x < kw; ++x) {
                        const int xi = w * sw - pw + x * dw;
                        if (xi < 0 || xi >= Wi) continue;
                        acc += ip[((long)zi * Hi + yi) * Wi + xi] *
                               wp[(z * kh + y) * kw + x];
                    }
                }
            }
        }
        out[((long)b * outCT + outC0 + o) * (long)Do * Ho * Wo +
            (long)d * Ho * Wo + h * Wo + w] = acc;
    }
}

// Per-channel batch-norm statistics over (B, D, H, W).
__global__ void bn_stats(const float* __restrict__ x, float* __restrict__ mean,
                         float* __restrict__ var, int CT, int c0, int S)
{
    const int c = blockIdx.x;
    __shared__ float s1[256], s2[256];
    float a = 0.f, q = 0.f;
    const long per = (long)Bv * S;
    for (long i = threadIdx.x; i < per; i += 256) {
        const long bb = i / S, sp = i % S;
        const float v = x[((long)bb * CT + c0 + c) * S + sp];
        a += v; q += v * v;
    }
    s1[threadIdx.x] = a; s2[threadIdx.x] = q;
    __syncthreads();
    for (int o = 128; o > 0; o >>= 1) {
        if (threadIdx.x < o) { s1[threadIdx.x] += s1[threadIdx.x + o];
                               s2[threadIdx.x] += s2[threadIdx.x + o]; }
        __syncthreads();
    }
    if (threadIdx.x == 0) {
        const float m = s1[0] / (float)per;
        mean[c] = m;
        var[c]  = s2[0] / (float)per - m * m;
    }
}

__global__ void bn_norm_relu(float* __restrict__ x, const float* __restrict__ g,
                             const float* __restrict__ bta,
                             const float* __restrict__ mean,
                             const float* __restrict__ var,
                             int CT, int c0, int Cn, int S, int doRelu)
{
    const long total = (long)Bv * Cn * S;
    for (long i = blockIdx.x * (long)blockDim.x + threadIdx.x; i < total;
         i += (long)gridDim.x * blockDim.x) {
        const long sp = i % S; long t = i / S;
        const int c = (int)(t % Cn); const int b = (int)(t / Cn);
        const long idx = ((long)b * CT + c0 + c) * S + sp;
        float v = (x[idx] - mean[c]) * rsqrtf(var[c] + EPSF) * g[c] + bta[c];
        x[idx] = doRelu ? fmaxf(v, 0.f) : v;
    }
}

// 3x3x3 max pool, stride 1, pad 1 on (Bv, CHm, Dv, Hv, Wvv).
__global__ void maxpool3_k(const float* __restrict__ in, float* __restrict__ out)
{
    const long total = (long)Bv * CHm * NPOS;
    for (long i = blockIdx.x * (long)blockDim.x + threadIdx.x; i < total;
         i += (long)gridDim.x * blockDim.x) {
        int w = (int)(i % Wvv); long t = i / Wvv;
        int h = (int)(t % Hv);  t /= Hv;
        int d = (int)(t % Dv);  t /= Dv;          // t = b*CHm + c
        const float* p = in + t * (long)NPOS;
        float m = -INFINITY;
        for (int dz = -1; dz <= 1; ++dz) {
            const int zz = d + dz; if (zz < 0 || zz >= Dv) continue;
            for (int dy = -1; dy <= 1; ++dy) {
                const int yy = h + dy; if (yy < 0 || yy >= Hv) continue;
                for (int dx = -1; dx <= 1; ++dx) {
                    const int xx = w + dx; if (xx < 0 || xx >= Wvv) continue;
                    m = fmaxf(m, p[((long)zz * Hv + yy) * Wvv + xx]);
                }
            }
        }
        out[i] = m;
    }
}

__global__ void copy_channels(const float* __restrict__ in, float* __restrict__ out,
                              int inCT, int inC0, int outCT, int outC0,
                              int Cn, int S)
{
    const long total = (long)Bv * Cn * S;
    for (long i = blockIdx.x * (long)blockDim.x + threadIdx.x; i < total;
         i += (long)gridDim.x * blockDim.x) {
        const long sp = i % S; long t = i / S;
        const int c = (int)(t % Cn); const int b = (int)(t / Cn);
        out[((long)b * outCT + outC0 + c) * S + sp] =
            in[((long)b * inCT + inC0 + c) * S + sp];
    }
}

__global__ void add_inplace(float* __restrict__ a, const float* __restrict__ b, long n)
{
    for (long i = blockIdx.x * (long)blockDim.x + threadIdx.x; i < n;
         i += (long)gridDim.x * blockDim.x)
        a[i] += b[i];
}

// Mean over spatial dims: one block per (b, c).
__global__ void mean_spatial(const float* __restrict__ in, float* __restrict__ out, int S)
{
    const int bc = blockIdx.x;
    __shared__ float sm[256];
    float a = 0.f;
    const float* p = in + (long)bc * S;
    for (int i = threadIdx.x; i < S; i += 256) a += p[i];
    sm[threadIdx.x] = a;
    __syncthreads();
    for (int o = 128; o > 0; o >>= 1) {
        if (threadIdx.x < o) sm[threadIdx.x] += sm[threadIdx.x + o];
        __syncthreads();
    }
    if (threadIdx.x == 0) out[bc] = sm[0] / (float)S;
}

// M = ym(8,16) @ lin_w(4096,16)^T + lin_b  -> (8,4096) == Mm(8,64,64)
__global__ void linear_k(const float* __restrict__ ym, const float* __restrict__ lw,
                         const float* __restrict__ lb, float* __restrict__ Mout)
{
    const long i = blockIdx.x * (long)blockDim.x + threadIdx.x;
    if (i >= (long)Bv * 4096) return;
    const int j = (int)(i % 4096); const int b = (int)(i / 4096);
    float a = lb[j];
#pragma unroll
    for (int k = 0; k < 16; ++k) a += ym[b * 16 + k] * lw[j * 16 + k];
    Mout[i] = a;
}

static inline int gblocks(long total)
{
    long b = (total + 255) / 256;
    return (int)(b > 16384 ? 16384 : b);
}

extern "C" void kernel_launch(void* const* d_in, const int* in_sizes, int n_in,
                              void* d_out, int out_size, void* d_ws, size_t ws_size,
                              hipStream_t stream)
{
    // ---- inputs (setup_inputs dict order: x, then params in insertion order)
    const float* x          = (const float*)d_in[0];
    const float* compress_w = (const float*)d_in[1];
    const float* expand_w   = (const float*)d_in[2];
    const float* t2_w       = (const float*)d_in[3];
    const float* t3_w       = (const float*)d_in[4];
    const float* s2_w       = (const float*)d_in[5];
    const float* s3_w       = (const float*)d_in[6];
    const float* nl_g       = (const float*)d_in[7];
    const float* nl_b       = (const float*)d_in[8];
    const float* maxproj_w  = (const float*)d_in[9];
    const float* c0_w       = (const float*)d_in[10];
    const float* bn0_g      = (const float*)d_in[11];
    const float* bn0_b      = (const float*)d_in[12];
    const float* c1_w  = (const float*)d_in[13]; const float* c1_b  = (const float*)d_in[14];
    const float* bn1_g = (const float*)d_in[15]; const float* bn1_b = (const float*)d_in[16];
    const float* c2_w  = (const float*)d_in[17]; const float* c2_b  = (const float*)d_in[18];
    const float* bn2_g = (const float*)d_in[19]; const float* bn2_b = (const float*)d_in[20];
    const float* c3_w  = (const float*)d_in[21]; const float* c3_b  = (const float*)d_in[22];
    const float* bn3_g = (const float*)d_in[23]; const float* bn3_b = (const float*)d_in[24];
    const float* c4_w  = (const float*)d_in[25]; const float* c4_b  = (const float*)d_in[26];
    const float* bn4_g = (const float*)d_in[27]; const float* bn4_b = (const float*)d_in[28];
    const float* c5_w  = (const float*)d_in[29]; const float* c5_b  = (const float*)d_in[30];
    const float* bn5_g = (const float*)d_in[31]; const float* bn5_b = (const float*)d_in[32];
    const float* c6_w  = (const float*)d_in[33]; const float* c6_b  = (const float*)d_in[34];
    const float* bn6_g = (const float*)d_in[35]; const float* bn6_b = (const float*)d_in[36];
    const float* lin_w = (const float*)d_in[37]; const float* lin_b = (const float*)d_in[38];

    // ---- outputs: (out, st2, Mm) concatenated flat
    const long OUTSZ = (long)Bv * Cx * NPOS;          // 25,690,112
    float* outPtr = (float*)d_out;
    float* st2Ptr = outPtr + OUTSZ;
    float* MmPtr  = st2Ptr + OUTSZ;                   // (8,64,64)

    // ---- workspace layout (floats)
    const long SFN = (long)Bv * CHm * NPOS;           // 6,422,528 per big buffer
    float* ws   = (float*)d_ws;
    float* buf0 = ws;                 // sf   -> later sf2
    float* buf1 = ws + SFN;           // pooled -> later qbuf/st2pre
    float* buf2 = ws + 2 * SFN;       // mx
    float* buf3 = ws + 3 * SFN;       // abuf -> st (in-place BN+relu)
    float* tb   = ws + 4 * SFN;       // tower scratch
    const long A0 = (long)Bv * 4 * NPOS;              // y0/y1/y2
    const long S3 = 9L * 15 * 15;                     // 2025
    const long A3 = (long)Bv * 8 * S3;                // y3/y4/y5
    const long S6 = 6L * 9 * 9;                       // 486
    const long A6 = (long)Bv * 16 * S6;               // y6
    float* y0 = tb;            float* y1 = y0 + A0;   float* y2 = y1 + A0;
    float* y3 = y2 + A0;       float* y4 = y3 + A3;   float* y5 = y4 + A3;
    float* y6 = y5 + A3;       float* ym = y6 + A6;   // (8,16)
    float* bmean = ym + 128;   float* bvar = bmean + 64;

    const dim3 blk(256);
    const dim3 gemmGrid((NPOS + 127) / 128, 1, Bv);
    const dim3 gemmGridG4((NPOS + 127) / 128, 4, Bv);

    // ---- 1) sf = grouped 1x1x1 compress(x): 4 groups of (16 out x 64 in)
    gemm_wmma_f32<64><<<gemmGridG4, blk, 0, stream>>>(
        x, compress_w, buf0, nullptr, nullptr,
        /*M=*/16, /*N=*/NPOS,
        /*xB=*/(long)Cx * NPOS, /*wB=*/0, /*yB=*/(long)CHm * NPOS,
        /*xG=*/64L * NPOS, /*wG=*/16L * 64, /*yG=*/16L * NPOS);

    // ---- 2) tower c0..c6 with per-conv BN+ReLU (tiny FLOPs, scalar kernels)
    conv3d_naive<<<gblocks(A0), blk, 0, stream>>>(buf0, c0_w, nullptr, y0,
        64, Dv, Hv, Wvv, 4, Dv, Hv, Wvv, 1,1,1, 1,1,1, 0,0,0, 1,1,1, 0,64, 0,4);
    bn_stats<<<4, blk, 0, stream>>>(y0, bmean, bvar, 4, 0, NPOS);
    bn_norm_relu<<<gblocks(A0), blk, 0, stream>>>(y0, bn0_g, bn0_b, bmean, bvar, 4,0,4, NPOS, 1);

    conv3d_naive<<<gblocks(A0), blk, 0, stream>>>(y0, c1_w, c1_b, y1,
        4, Dv, Hv, Wvv, 4, Dv, Hv, Wvv, 3,3,3, 1,1,1, 1,1,1, 1,1,1, 0,4, 0,4);
    bn_stats<<<4, blk, 0, stream>>>(y1, bmean, bvar, 4, 0, NPOS);
    bn_norm_relu<<<gblocks(A0), blk, 0, stream>>>(y1, bn1_g, bn1_b, bmean, bvar, 4,0,4, NPOS, 1);

    conv3d_naive<<<gblocks(A0), blk, 0, stream>>>(y1, c2_w, c2_b, y2,
        4, Dv, Hv, Wvv, 4, Dv, Hv, Wvv, 3,3,3, 1,1,1, 1,1,1, 1,1,1, 0,4, 0,4);
    bn_stats<<<4, blk, 0, stream>>>(y2, bmean, bvar, 4, 0, NPOS);
    bn_norm_relu<<<gblocks(A0), blk, 0, stream>>>(y2, bn2_g, bn2_b, bmean, bvar, 4,0,4, NPOS, 1);

    conv3d_naive<<<gblocks(A3), blk, 0, stream>>>(y2, c3_w, c3_b, y3,
        4, Dv, Hv, Wvv, 8, 9, 15, 15, 3,3,3, 2,2,2, 2,2,2, 1,1,1, 0,4, 0,8);
    bn_stats<<<8, blk, 0, stream>>>(y3, bmean, bvar, 8, 0, (int)S3);
    bn_norm_relu<<<gblocks(A3), blk, 0, stream>>>(y3, bn3_g, bn3_b, bmean, bvar, 8,0,8, (int)S3, 1);

    conv3d_naive<<<gblocks(A3), blk, 0, stream>>>(y3, c4_w, c4_b, y4,
        8, 9, 15, 15, 8, 9, 15, 15, 3,3,3, 1,1,1, 1,1,1, 1,1,1, 0,8, 0,8);
    bn_stats<<<8, blk, 0, stream>>>(y4, bmean, bvar, 8, 0, (int)S3);
    bn_norm_relu<<<gblocks(A3), blk, 0, stream>>>(y4, bn4_g, bn4_b, bmean, bvar, 8,0,8, (int)S3, 1);

    conv3d_naive<<<gblocks(A3), blk, 0, stream>>>(y4, c5_w, c5_b, y5,
        8, 9, 15, 15, 8, 9, 15, 15, 3,3,3, 1,1,1, 1,1,1, 1,1,1, 0,8, 0,8);
    bn_stats<<<8, blk, 0, stream>>>(y5, bmean, bvar, 8, 0, (int)S3);
    bn_norm_relu<<<gblocks(A3), blk, 0, stream>>>(y5, bn5_g, bn5_b, bmean, bvar, 8,0,8, (int)S3, 1);

    conv3d_naive<<<gblocks(A6), blk, 0, stream>>>(y5, c6_w, c6_b, y6,
        8, 9, 15, 15, 16, 6, 9, 9, 3,3,3, 2,2,2, 2,2,2, 1,1,1, 0,8, 0,16);
    bn_stats<<<16, blk, 0, stream>>>(y6, bmean, bvar, 16, 0, (int)S6);
    bn_norm_relu<<<gblocks(A6), blk, 0, stream>>>(y6, bn6_g, bn6_b, bmean, bvar, 16,0,16, (int)S6, 1);

    mean_spatial<<<Bv * 16, blk, 0, stream>>>(y6, ym, (int)S6);
    linear_k<<<(Bv * 4096 + 255) / 256, blk, 0, stream>>>(ym, lin_w, lin_b, MmPtr);

    // ---- 3) mx = maxproj(maxpool3(sf))   [64x64 WMMA GEMM]
    maxpool3_k<<<gblocks(SFN), blk, 0, stream>>>(buf0, buf1);
    gemm_wmma_f32<64><<<gemmGrid, blk, 0, stream>>>(
        buf1, maxproj_w, buf2, nullptr, nullptr,
        64, NPOS, (long)CHm * NPOS, 0, (long)CHm * NPOS, 0, 0, 0);

    // ---- 4) a-branch -> st = relu(bn(concat(a1, a2, a3)))
    copy_channels<<<gblocks((long)Bv * 21 * NPOS), blk, 0, stream>>>(
        buf0, buf3, 64, 0, 64, 0, 21, NPOS);
    conv3d_naive<<<gblocks((long)Bv * 21 * NPOS), blk, 0, stream>>>(buf0, t2_w, nullptr, buf3,
        21, Dv, Hv, Wvv, 21, Dv, Hv, Wvv, 3,1,1, 1,1,1, 1,0,0, 1,1,1, 21,64, 21,64);
    conv3d_naive<<<gblocks((long)Bv * 22 * NPOS), blk, 0, stream>>>(buf0, t3_w, nullptr, buf3,
        22, Dv, Hv, Wvv, 22, Dv, Hv, Wvv, 3,1,1, 1,1,1, 2,0,0, 2,1,1, 42,64, 42,64);
    bn_stats<<<64, blk, 0, stream>>>(buf3, bmean, bvar, 64, 0, NPOS);
    bn_norm_relu<<<gblocks(SFN), blk, 0, stream>>>(buf3, nl_g, nl_b, bmean, bvar, 64,0,64, NPOS, 1);

    // ---- 5) sf2 = einsum('bcthw,boc->bothw', st, Mm): per-batch 64x64 WMMA
    gemm_wmma_f32<64><<<gemmGrid, blk, 0, stream>>>(
        buf3, MmPtr, buf0, nullptr, nullptr,
        64, NPOS, (long)CHm * NPOS, 4096, (long)CHm * NPOS, 0, 0, 0);

    // ---- 6) q-branch -> st2pre = concat(q1, q2, q3) + mx   (into buf1)
    copy_channels<<<gblocks((long)Bv * 21 * NPOS), blk, 0, stream>>>(
        buf0, buf1, 64, 0, 64, 0, 21, NPOS);
    conv3d_naive<<<gblocks((long)Bv * 21 * NPOS), blk, 0, stream>>>(buf0, s2_w, nullptr, buf1,
        21, Dv, Hv, Wvv, 21, Dv, Hv, Wvv, 1,3,3, 1,1,1, 0,1,1, 1,1,1, 21,64, 21,64);
    conv3d_naive<<<gblocks((long)Bv * 22 * NPOS), blk, 0, stream>>>(buf0, s3_w, nullptr, buf1,
        22, Dv, Hv, Wvv, 22, Dv, Hv, Wvv, 1,3,3, 1,1,1, 0,2,2, 1,2,2, 42,64, 42,64);
    add_inplace<<<gblocks(SFN), blk, 0, stream>>>(buf1, buf2, SFN);

    // ---- 7) st2 = grouped expand(st2pre); out = st2 + x  (fused epilogue)
    gemm_wmma_f32<16><<<gemmGridG4, blk, 0, stream>>>(
        buf1, expand_w, st2Ptr, outPtr, x,
        /*M=*/64, /*N=*/NPOS,
        /*xB=*/(long)CHm * NPOS, /*wB=*/0, /*yB=*/(long)Cx * NPOS,
        /*xG=*/16L * NPOS, /*wG=*/64L * 16, /*yG=*/64L * NPOS);
}